// GNN_48619029791117
// MI455X (gfx1250) — compile-verified
//
#include <hip/hip_runtime.h>

#define NN 100000      // nodes
#define NE 1600000     // edges
#define NG 2048        // graphs
#define HID 100        // hidden width
#define C0  33         // input features
#define C0P 36         // padded input features (mult of 4)
#define NPAD 112       // padded column count (7 * 16)
#define NT_TILES 7     // ceil(100/16)
#define NM_TILES (NN/16)  // 6250 exactly

typedef __attribute__((ext_vector_type(2))) float v2f;
typedef __attribute__((ext_vector_type(8))) float v8f;

// ---------------- utility kernels ----------------

__global__ __launch_bounds__(256) void gnn_fill(float* __restrict__ p, int n, float v) {
    int i = blockIdx.x * blockDim.x + threadIdx.x;
    if (i < n) p[i] = v;
}

// deg: atomic count of destinations (deg buffer pre-filled with 1.0 for self loops)
__global__ __launch_bounds__(256) void gnn_deg(const int* __restrict__ ei, float* __restrict__ deg) {
    int e = blockIdx.x * blockDim.x + threadIdx.x;
    if (e < NE) atomicAdd(&deg[ei[NE + e]], 1.0f);
}

__global__ __launch_bounds__(256) void gnn_rsqrt(float* __restrict__ d, int n) {
    int i = blockIdx.x * blockDim.x + threadIdx.x;
    if (i < n) d[i] = rsqrtf(d[i]);   // deg >= 1 always (self loop)
}

__global__ __launch_bounds__(256) void gnn_norm(const int* __restrict__ ei,
                                                const float* __restrict__ dis,
                                                float* __restrict__ nrm) {
    int e = blockIdx.x * blockDim.x + threadIdx.x;
    if (e < NE) nrm[e] = dis[ei[e]] * dis[ei[NE + e]];
}

// fold bias + BN(eval) into per-column scale/shift for all 4 layers (400 threads)
__global__ __launch_bounds__(256) void gnn_scaleshift(const float* __restrict__ bias,
                                                      const float* __restrict__ gamma,
                                                      const float* __restrict__ beta,
                                                      const float* __restrict__ mean,
                                                      const float* __restrict__ var,
                                                      float* __restrict__ scale,
                                                      float* __restrict__ shift) {
    int i = blockIdx.x * blockDim.x + threadIdx.x;
    if (i < 4 * HID) {
        float rs = rsqrtf(var[i] + 1e-5f);
        float s  = gamma[i] * rs;
        scale[i] = s;
        shift[i] = (bias[i] - mean[i]) * s + beta[i];
    }
}

// zero-pad x [NN,33] -> xp [NN,36]
__global__ __launch_bounds__(256) void gnn_padX(const float* __restrict__ x, float* __restrict__ xp) {
    int idx = blockIdx.x * blockDim.x + threadIdx.x;
    if (idx >= NN * C0P) return;
    int i = idx / C0P;
    int c = idx - i * C0P;
    xp[idx] = (c < C0) ? x[(size_t)i * C0 + c] : 0.f;
}

// pack W [K,100] into WMMA-swizzled, zero-padded Bs: for k2 in [0,K2), n in [0,112):
//   Bs[(k2*112 + n)*2 + {0,1}] = (W[2k2][n], W[2k2+1][n])   (0 beyond K or col>=100)
__global__ __launch_bounds__(256) void gnn_packB(const float* __restrict__ W, int K, int K2,
                                                 float* __restrict__ Bs) {
    int idx = blockIdx.x * blockDim.x + threadIdx.x;
    if (idx >= K2 * NPAD) return;
    int k2 = idx / NPAD;
    int n  = idx - k2 * NPAD;
    int k  = k2 * 2;
    float lo = (k     < K && n < HID) ? W[(size_t)k * HID + n]       : 0.f;
    float hi = (k + 1 < K && n < HID) ? W[(size_t)(k + 1) * HID + n] : 0.f;
    Bs[(size_t)idx * 2 + 0] = lo;
    Bs[(size_t)idx * 2 + 1] = hi;
}

// ---------------- WMMA GEMM: Y[NN,100] = X[NN,KEFF] @ W ----------------
// one wave = one 16x16 output tile; branch-free inner loop:
//   1x global_load_b64 (A pair) + 1x global_load_b64 (packed B pair) + v_wmma_f32_16x16x4_f32
template<int KEFF, int LDA>
__global__ __launch_bounds__(256) void gnn_gemm_wmma(const float* __restrict__ X,
                                                     const float* __restrict__ Bs,
                                                     float* __restrict__ Y) {
    int wave = (blockIdx.x * blockDim.x + threadIdx.x) >> 5;   // wave32
    int lane = threadIdx.x & 31;
    if (wave >= NM_TILES * NT_TILES) return;                   // wave-uniform exit

    int tileM = wave / NT_TILES;
    int tileN = wave - tileM * NT_TILES;
    int m0 = tileM << 4;
    int n0 = tileN << 4;

    int half = lane >> 4;        // selects K pair {k0+2h, k0+2h+1}
    int p    = lane & 15;        // A: row M; B: col N

    const float* ap = X  + (size_t)(m0 + p) * LDA + 2 * half;          // 8B aligned
    const float* bp = Bs + ((size_t)half * NPAD + (n0 + p)) * 2;       // 8B aligned

    v8f c = {0.f, 0.f, 0.f, 0.f, 0.f, 0.f, 0.f, 0.f};
    #pragma unroll
    for (int k0 = 0; k0 < KEFF; k0 += 4) {
        v2f a = *(const v2f*)ap;     // X[m][k0+2h], X[m][k0+2h+1]
        v2f b = *(const v2f*)bp;     // W[k0+2h][n], W[k0+2h+1][n]
        ap += 4;
        bp += 2 * 2 * NPAD;          // advance two k2-rows (k += 4)
        c = __builtin_amdgcn_wmma_f32_16x16x4_f32(false, a, false, b,
                                                  (short)0, c, false, false);
    }

    int n = n0 + p;
    if (n < HID) {
        #pragma unroll
        for (int r = 0; r < 8; ++r)
            Y[(size_t)(m0 + r + (half << 3)) * HID + n] = c[r];
    }
}

// ---------------- edge aggregation: agg[col] += xw[row] * norm ----------------
__global__ __launch_bounds__(256) void gnn_agg(const int* __restrict__ ei,
                                               const float* __restrict__ nrm,
                                               const float* __restrict__ xw,
                                               float* __restrict__ agg) {
    int idx = blockIdx.x * blockDim.x + threadIdx.x;
    if (idx >= NE * 25) return;
    int e = idx / 25;
    int c = (idx - e * 25) << 2;
    int r  = ei[e];
    int cl = ei[NE + e];
    float nv = nrm[e];
    const float4 v = *(const float4*)(xw + (size_t)r * HID + c);
    float* dst = agg + (size_t)cl * HID + c;
    atomicAdd(dst + 0, v.x * nv);
    atomicAdd(dst + 1, v.y * nv);
    atomicAdd(dst + 2, v.z * nv);
    atomicAdd(dst + 3, v.w * nv);
}

// ---------------- fused self-loop + BN(+ReLU), in place on agg buffer ----------------
__global__ __launch_bounds__(256) void gnn_bnrelu(float* __restrict__ h,
                                                  const float* __restrict__ xw,
                                                  const float* __restrict__ dis,
                                                  const float* __restrict__ scale,
                                                  const float* __restrict__ shift,
                                                  int relu) {
    int idx = blockIdx.x * blockDim.x + threadIdx.x;
    if (idx >= NN * 25) return;
    int i = idx / 25;
    int c = (idx - i * 25) << 2;
    float d2 = dis[i] * dis[i];
    float4 a = *(float4*)(h + (size_t)i * HID + c);
    const float4 w = *(const float4*)(xw + (size_t)i * HID + c);
    float t0 = (a.x + w.x * d2) * scale[c + 0] + shift[c + 0];
    float t1 = (a.y + w.y * d2) * scale[c + 1] + shift[c + 1];
    float t2 = (a.z + w.z * d2) * scale[c + 2] + shift[c + 2];
    float t3 = (a.w + w.w * d2) * scale[c + 3] + shift[c + 3];
    if (relu) {
        t0 = fmaxf(t0, 0.f); t1 = fmaxf(t1, 0.f);
        t2 = fmaxf(t2, 0.f); t3 = fmaxf(t3, 0.f);
    }
    *(float4*)(h + (size_t)i * HID + c) = make_float4(t0, t1, t2, t3);
}

// ---------------- global add pool ----------------
__global__ __launch_bounds__(256) void gnn_pool(const float* __restrict__ h,
                                                const int* __restrict__ batch,
                                                float* __restrict__ G) {
    int idx = blockIdx.x * blockDim.x + threadIdx.x;
    if (idx >= NN * 25) return;
    int i = idx / 25;
    int c = (idx - i * 25) << 2;
    int g = batch[i];
    const float4 v = *(const float4*)(h + (size_t)i * HID + c);
    float* dst = G + (size_t)g * HID + c;
    atomicAdd(dst + 0, v.x);
    atomicAdd(dst + 1, v.y);
    atomicAdd(dst + 2, v.z);
    atomicAdd(dst + 3, v.w);
}

// ---------------- head: out = LeakyReLU(G @ head_W + b), one wave per graph ----------------
__global__ __launch_bounds__(256) void gnn_head(const float* __restrict__ G,
                                                const float* __restrict__ hw,
                                                const float* __restrict__ hb,
                                                float* __restrict__ out) {
    int wave = (blockIdx.x * blockDim.x + threadIdx.x) >> 5;
    int lane = threadIdx.x & 31;
    if (wave >= NG) return;
    float s = 0.f;
    for (int c = lane; c < HID; c += 32)
        s += G[(size_t)wave * HID + c] * hw[c];
    #pragma unroll
    for (int off = 16; off >= 1; off >>= 1)
        s += __shfl_xor(s, off, 32);
    if (lane == 0) {
        float t = s + hb[0];
        out[wave] = (t >= 0.f) ? t : 0.1f * t;
    }
}

// ---------------- launcher ----------------
extern "C" void kernel_launch(void* const* d_in, const int* in_sizes, int n_in,
                              void* d_out, int out_size, void* d_ws, size_t ws_size,
                              hipStream_t stream) {
    const float* x      = (const float*)d_in[0];
    const int*   ei     = (const int*)d_in[1];   // [2, NE]
    const int*   batch  = (const int*)d_in[2];   // [NN]
    const float* W0     = (const float*)d_in[3];
    const float* Ws     = (const float*)d_in[4];
    const float* biases = (const float*)d_in[5];
    const float* gamma  = (const float*)d_in[6];
    const float* beta   = (const float*)d_in[7];
    const float* bnm    = (const float*)d_in[8];
    const float* bnv    = (const float*)d_in[9];
    const float* headW  = (const float*)d_in[10];
    const float* headB  = (const float*)d_in[11];
    float* out = (float*)d_out;

    float* ws = (float*)d_ws;
    float* bufA  = ws;                          // [NN,100] agg / h
    float* bufB  = bufA + (size_t)NN * HID;     // [NN,100] xw
    float* bufX  = bufB + (size_t)NN * HID;     // [NN,36] padded input
    float* dis   = bufX + (size_t)NN * C0P;     // [NN]
    float* nrm   = dis + NN;                    // [NE]
    float* scale = nrm + NE;                    // [4,100]
    float* shift = scale + 4 * HID;             // [4,100]
    float* Bs    = shift + 4 * HID;             // [50,112] float2 (packed W)
    float* G     = Bs + (size_t)50 * NPAD * 2;  // [NG,100]

    const int B = 256;
    auto nb = [](long long n, int b) { return (int)((n + b - 1) / b); };

    // degree / normalization / BN folding / padded input
    gnn_fill <<<nb(NN, B), B, 0, stream>>>(dis, NN, 1.0f);      // self loop
    gnn_deg  <<<nb(NE, B), B, 0, stream>>>(ei, dis);
    gnn_rsqrt<<<nb(NN, B), B, 0, stream>>>(dis, NN);
    gnn_norm <<<nb(NE, B), B, 0, stream>>>(ei, dis, nrm);
    gnn_scaleshift<<<nb(4 * HID, B), B, 0, stream>>>(biases, gamma, beta, bnm, bnv, scale, shift);
    gnn_padX <<<nb((long long)NN * C0P, B), B, 0, stream>>>(x, bufX);

    const int gemm_blocks = nb((long long)NM_TILES * NT_TILES * 32, B);
    for (int l = 0; l < 4; ++l) {
        int K            = (l == 0) ? C0 : HID;
        int K2           = (l == 0) ? (C0P / 2) : (HID / 2);
        const float* Wl  = (l == 0) ? W0 : (Ws + (size_t)(l - 1) * HID * HID);

        gnn_packB<<<nb((long long)K2 * NPAD, B), B, 0, stream>>>(Wl, K, K2, Bs);
        if (l == 0)
            gnn_gemm_wmma<C0P, C0P><<<gemm_blocks, B, 0, stream>>>(bufX, Bs, bufB);
        else
            gnn_gemm_wmma<HID, HID><<<gemm_blocks, B, 0, stream>>>(bufA, Bs, bufB);

        gnn_fill<<<nb((long long)NN * HID, B), B, 0, stream>>>(bufA, NN * HID, 0.0f);
        gnn_agg <<<nb((long long)NE * 25, B), B, 0, stream>>>(ei, nrm, bufB, bufA);
        gnn_bnrelu<<<nb((long long)NN * 25, B), B, 0, stream>>>(bufA, bufB, dis,
                                                                scale + l * HID, shift + l * HID,
                                                                (l < 3) ? 1 : 0);
    }

    gnn_fill<<<nb((long long)NG * HID, B), B, 0, stream>>>(G, NG * HID, 0.0f);
    gnn_pool<<<nb((long long)NN * 25, B), B, 0, stream>>>(bufA, batch, G);
    gnn_head<<<nb((long long)NG * 32, B), B, 0, stream>>>(G, headW, headB, out);
}